// NeuralODEPlasticity_64209761075274
// MI455X (gfx1250) — compile-verified
//
#include <hip/hip_runtime.h>

typedef float v2f __attribute__((ext_vector_type(2)));
typedef float v8f __attribute__((ext_vector_type(8)));

// CDNA5 hardware transcendental TANH (v_tanh_f32).
__device__ __forceinline__ float fast_tanhf(float x) {
#if __has_builtin(__builtin_amdgcn_tanhf)
    return __builtin_amdgcn_tanhf(x);
#else
    float r;
    asm("v_tanh_f32 %0, %1" : "=v"(r) : "v"(x));
    return r;
#endif
}

// One evaluation of ode_fn for 16 elements held in lanes 0..15 (A.x = state,
// zero elsewhere). Returns dy for element m in lane m (m = 0..15); lanes
// 16..31 return garbage (masked by caller).
__device__ __forceinline__ float ode_eval(float s,
                                          const v2f Bops[4],
                                          const v8f Cops[4],
                                          const float w2m[4],
                                          float bias, int lane) {
    v2f A;
    A.x = s;      // lanes 0..15: K=0 column of A; caller guarantees 0 elsewhere
    A.y = 0.0f;   // K=1 (and K=2,3 via lanes 16..31) are zero

    float acc[8];
#pragma unroll
    for (int v = 0; v < 8; ++v) acc[v] = 0.0f;

#pragma unroll
    for (int t = 0; t < 4; ++t) {
        // D = A x B + C : D[m][n] = s_m * a_n + c_n  (16x16 tile, n-slice t)
        v8f d = __builtin_amdgcn_wmma_f32_16x16x4_f32(
            /*neg_a=*/false, A, /*neg_b=*/false, Bops[t],
            /*c_mod=*/(short)0, Cops[t], /*reuse_a=*/false, /*reuse_b=*/false);
#pragma unroll
        for (int v = 0; v < 8; ++v)
            acc[v] = fmaf(fast_tanhf(d[v]), w2m[t], acc[v]);
    }

    // Transpose-fused butterfly row-sum. acc[v] at lane L is the column
    // contribution (col = L&15) to row v (+8 for lanes 16..31). At xor-round
    // k, lane bit k selects the row bit while the shuffle halves the lane
    // span: after masks 1,2,4 each lane holds the 8-lane partial of row L&7;
    // mask 8 completes the 16-lane sum within each half.
    const bool b0 = (lane & 1) != 0;
    const bool b1 = (lane & 2) != 0;
    const bool b2 = (lane & 4) != 0;

    float t0[4];
#pragma unroll
    for (int p = 0; p < 4; ++p) {
        const float own = b0 ? acc[2 * p + 1] : acc[2 * p];
        const float oth = b0 ? acc[2 * p]     : acc[2 * p + 1];
        t0[p] = own + __shfl_xor(oth, 1, 32);
    }
    float t1[2];
#pragma unroll
    for (int q = 0; q < 2; ++q) {
        const float own = b1 ? t0[2 * q + 1] : t0[2 * q];
        const float oth = b1 ? t0[2 * q]     : t0[2 * q + 1];
        t1[q] = own + __shfl_xor(oth, 2, 32);
    }
    const float own = b2 ? t1[1] : t1[0];
    const float oth = b2 ? t1[0] : t1[1];
    float w = own + __shfl_xor(oth, 4, 32);
    w += __shfl_xor(w, 8, 32);          // lane L: dy[(L&7) + 8*(L>=16)]
    const float o = __shfl_xor(w, 16, 32);
    // Lane m (0..15): m<8 -> own half (dy[m]); m>=8 -> other half (dy[m]).
    return ((lane & 8) ? o : w) + bias;
}

__global__ __launch_bounds__(256) void node_rk4_wmma(
    const float* __restrict__ y_in, const float* __restrict__ ctx,
    const float* __restrict__ meta_p, const float* __restrict__ W1,
    const float* __restrict__ b1, const float* __restrict__ W2,
    const float* __restrict__ b2, const float* __restrict__ dt_p,
    float* __restrict__ y_out, int n_elems) {
    const int lane = threadIdx.x & 31;
    const int wave = (int)((blockIdx.x * blockDim.x + threadIdx.x) >> 5);
    const int base = wave * 16;
    const bool lo  = lane < 16;
    const int  nl  = lane & 15;

    const float meta = meta_p[0];
    const float h    = dt_p[0] * 0.25f;   // dt / N_STEPS
    const float bias = b2[0] * meta;      // b2 * meta, added once per eval
    const float h2   = 0.5f * h;
    const float h6   = h * (1.0f / 6.0f);

    float cx[8];
#pragma unroll
    for (int k = 0; k < 8; ++k) cx[k] = ctx[k];

    // Fold ctx @ W1[1:, :] + b1 into the WMMA C operand; W1[0, :] becomes the
    // (only) nonzero row of B; meta folds into w2.
    v2f Bops[4];
    v8f Cops[4];
    float w2m[4];
#pragma unroll
    for (int t = 0; t < 4; ++t) {
        const int j = t * 16 + nl;          // hidden unit index for this lane
        const float a = W1[j];              // W1[0, j]
        float c = b1[j];
#pragma unroll
        for (int k = 0; k < 8; ++k) c = fmaf(cx[k], W1[(k + 1) * 64 + j], c);
        Bops[t].x = lo ? a : 0.0f;          // K=0 row of B; rest zero
        Bops[t].y = 0.0f;
#pragma unroll
        for (int v = 0; v < 8; ++v) Cops[t][v] = c;  // C[m][n] = c_n, all m
        w2m[t] = W2[j] * meta;
    }

    const int idx = base + lane;
    float y = 0.0f;
    if (lo && idx < n_elems) y = y_in[idx];

    for (int step = 0; step < 4; ++step) {
        const float k1 = ode_eval(y, Bops, Cops, w2m, bias, lane);
        const float s2 = lo ? fmaf(h2, k1, y) : 0.0f;
        const float k2 = ode_eval(s2, Bops, Cops, w2m, bias, lane);
        const float s3 = lo ? fmaf(h2, k2, y) : 0.0f;
        const float k3 = ode_eval(s3, Bops, Cops, w2m, bias, lane);
        const float s4 = lo ? fmaf(h, k3, y) : 0.0f;
        const float k4 = ode_eval(s4, Bops, Cops, w2m, bias, lane);
        const float upd = k1 + 2.0f * k2 + 2.0f * k3 + k4;
        y = lo ? fmaf(h6, upd, y) : 0.0f;
    }

    if (lo && idx < n_elems) y_out[idx] = y;
}

extern "C" void kernel_launch(void* const* d_in, const int* in_sizes, int n_in,
                              void* d_out, int out_size, void* d_ws, size_t ws_size,
                              hipStream_t stream) {
    const float* y_in = (const float*)d_in[0];
    const float* ctx  = (const float*)d_in[1];
    const float* meta = (const float*)d_in[2];
    const float* W1   = (const float*)d_in[3];
    const float* b1   = (const float*)d_in[4];
    const float* W2   = (const float*)d_in[5];
    const float* b2   = (const float*)d_in[6];
    const float* dt   = (const float*)d_in[7];
    float* out = (float*)d_out;

    const int n = in_sizes[0];                 // 512*4096 elements
    const int waves = (n + 15) / 16;           // 16 elements per wave
    const int blocks = (waves + 7) / 8;        // 8 waves (256 threads) per block

    node_rk4_wmma<<<blocks, 256, 0, stream>>>(y_in, ctx, meta, W1, b1, W2, b2,
                                              dt, out, n);
}